// DiffKin_49546742727280
// MI455X (gfx1250) — compile-verified
//
#include <hip/hip_runtime.h>
#include <math.h>

// Forward kinematics for MI455X (gfx1250, wave32).
// One thread per batch element; 64-frame chain kept in registers.
// CDNA5-specific paths used:
//   * tensor_load_to_lds (TDM) stages the shared origins (64x4x4 f32 = 4KB)
//     into LDS per block, descriptor built per ISA cdna5 §8.3/8.4.
//   * s_wait_tensorcnt 0 before consuming the staged data.
//   * global_prefetch_b8 (via __builtin_prefetch) warms the angle slab.
//   * hardware transcendentals (v_sin/v_cos/v_sqrt/v_rcp) keep the VALU side
//     under the 11.5us HBM store roofline (256MB @ 23.3TB/s).
// WMMA is intentionally not used: the matrix ops are sequentially-dependent
// batched 4x4 affine composes (both operands batch-varying); the best WMMA
// embedding is 25%-utilized plus layout shuffles - slower than per-lane FMAs.

#define NF_MAX 64
#define BT     128
#define EPS_F  1e-8f

typedef unsigned int uint32x4 __attribute__((ext_vector_type(4)));
typedef int          int32x8  __attribute__((ext_vector_type(8)));
typedef int          int32x4  __attribute__((ext_vector_type(4)));

__global__ __launch_bounds__(BT) void fk_chain_kernel(
    const float* __restrict__ JA,       // [B, nCtrl] joint angles
    const float* __restrict__ axes,     // [nF, 3]
    const float* __restrict__ origins,  // [nF, 4, 4]
    const float* __restrict__ mmul,     // [nMimic]
    const float* __restrict__ moff,     // [nMimic]
    const int*   __restrict__ ctrl,     // [nCtrl]
    const int*   __restrict__ mdst,     // [nMimic]
    const int*   __restrict__ msrc,     // [nMimic]
    const int*   __restrict__ jtype,    // [nF]
    float* __restrict__ out,            // [B, nF, 4, 4]
    int B, int nF, int nCtrl, int nMimic)
{
    __shared__ float  sOrig [NF_MAX * 16];  // filled by TDM (row-major 4x4 per frame)
    __shared__ float4 sDescA[NF_MAX];       // {ax, ay, az, mult}
    __shared__ float4 sDescB[NF_MAX];       // {off, rev, pri, src_as_bits}
    __shared__ float  sAng  [NF_MAX * BT];  // angles transposed: [col][lane]

    const int tid       = threadIdx.x;
    const int blockBase = blockIdx.x * BT;
    const int gb        = blockBase + tid;

    // ---- Tensor Data Mover: async stage origins (nF*16 f32) into LDS -------
    // Issued by wave 0 only; D# built per CDNA5 ISA §8.3 (group0) / §8.4 (group1).
    if (tid < 32) {
        unsigned long long ga  = (unsigned long long)(uintptr_t)origins;
        unsigned int ldsOff    = (unsigned int)(uintptr_t)(&sOrig[0]); // low 32b of flat LDS addr = LDS offset
        unsigned int nElem     = (unsigned int)(nF * 16);              // f32 elements (data_size units)
        uint32x4 g0 = {
            1u,                                              // count=1 (valid), user mode, no gather
            ldsOff,                                          // lds_addr [63:32]
            (unsigned int)(ga & 0xffffffffull),              // global_addr[31:0]
            (unsigned int)((ga >> 32) & 0x1ffffffull)        // global_addr[56:32]
                | (2u << 30)                                 // type=2 ("image")
        };
        int32x8 g1 = {
            (int)(2u << 16),                                 // data_size=2 (4 bytes); mask/flags/pad = 0
            (int)((nElem & 0xffffu) << 16),                  // tensor_dim0[15:0]  @ bits 63:48
            (int)(((nElem >> 16) & 0xffffu) | (1u << 16)),   // tensor_dim0[31:16] | tensor_dim1=1 (lo)
            (int)((nElem & 0xffffu) << 16),                  // tensor_dim1(hi)=0 | tile_dim0 = nElem
            1,                                               // tile_dim1=1, tile_dim2=0
            (int)nElem,                                      // tensor_dim0_stride[31:0]
            0,                                               // stride hi / dim1_stride lo = 0
            0
        };
        int32x4 zz4 = {0, 0, 0, 0};
        int32x8 zz8 = {0, 0, 0, 0, 0, 0, 0, 0};
        // amdgpu-toolchain (clang-23) 6-arg form: (g0, g1, g2, g3, g4, cpol)
        __builtin_amdgcn_tensor_load_to_lds(g0, g1, zz4, zz4, zz8, 0); // TENSORcnt++
    }

    // ---- Prefetch this block's angle rows (emits global_prefetch_b8) -------
    if (gb < B) __builtin_prefetch(&JA[(size_t)gb * nCtrl], 0, 0);

    // ---- Build per-frame descriptors (scatter ctrlable, then mimic) --------
    if (tid < nF) {
        int jt = jtype[tid];
        sDescA[tid] = make_float4(axes[3 * tid + 0], axes[3 * tid + 1],
                                  axes[3 * tid + 2], 1.0f);
        sDescB[tid] = make_float4(0.0f,
                                  (jt == 1) ? 1.0f : 0.0f,
                                  (jt == 2) ? 1.0f : 0.0f,
                                  __int_as_float(-1));
    }
    __syncthreads();
    if (tid < nCtrl) sDescB[ctrl[tid]].w = __int_as_float(tid);   // frame -> JA column
    __syncthreads();
    bool  dm = (tid < nMimic);
    int   mD = 0; float mS = 0.f, mM = 0.f, mO = 0.f;
    if (dm) { mD = mdst[tid]; mS = sDescB[msrc[tid]].w; mM = mmul[tid]; mO = moff[tid]; }
    __syncthreads();                            // read post-scatter state first
    if (dm) { sDescB[mD].w = mS; sDescA[mD].w = mM; sDescB[mD].x = mO; }

    // ---- Stage this block's angles transposed into LDS ---------------------
    for (int col = 0; col < nCtrl; ++col)
        sAng[col * BT + tid] = (gb < B) ? JA[(size_t)gb * nCtrl + col] : 0.0f;

    if (tid < 32) __builtin_amdgcn_s_wait_tensorcnt(0);  // TDM data landed
    __syncthreads();

    if (gb >= B) return;

    // ---- Sequential chain: pose (3x4 affine) in registers -------------------
    float p00 = 1.f, p01 = 0.f, p02 = 0.f, p03 = 0.f;
    float p10 = 0.f, p11 = 1.f, p12 = 0.f, p13 = 0.f;
    float p20 = 0.f, p21 = 0.f, p22 = 1.f, p23 = 0.f;
    float* outB = out + (size_t)gb * nF * 16;

    #pragma unroll 1
    for (int n = 0; n < nF; ++n) {
        float4 da = sDescA[n];                   // ax, ay, az, mult
        float4 db = sDescB[n];                   // off, rev, pri, src
        int    src = __float_as_int(db.w);
        float  a0  = (src >= 0) ? sAng[src * BT + tid] : 0.0f;
        float  ang = fmaf(a0, da.w, db.x);

        float ax = da.x, ay = da.y, az = da.z;
        float ar = ang * db.y;                   // revolute component
        float ap = ang * db.z;                   // prismatic component
        float wx = ax * ar, wy = ay * ar, wz = az * ar;

        // Rodrigues: R = ct*I + s*K + c*w w^T,  ct = 1 - c*th2
        // Hardware transcendentals: v_sqrt / v_sin / v_cos / v_rcp.
        float th2  = wx * wx + wy * wy + wz * wz;
        float th   = __builtin_amdgcn_sqrtf(th2);
        bool  tiny = th < EPS_F;
        float safe = tiny ? 1.0f : th;
        float sn   = __sinf(safe);
        float cs   = __cosf(safe);
        float inv  = __builtin_amdgcn_rcpf(safe);
        float s    = tiny ? 1.0f : sn * inv;
        float c    = tiny ? 0.5f : (1.0f - cs) * inv * inv;
        float ct   = 1.0f - c * th2;

        float cwx = c * wx, cwy = c * wy, cwz = c * wz;
        float swx = s * wx, swy = s * wy, swz = s * wz;
        float r00 = ct + cwx * wx, r01 = cwx * wy - swz, r02 = cwx * wz + swy;
        float r10 = cwx * wy + swz, r11 = ct + cwy * wy, r12 = cwy * wz - swx;
        float r20 = cwx * wz - swy, r21 = cwy * wz + swx, r22 = ct + cwz * wz;
        float tx = ax * ap, ty = ay * ap, tz = az * ap;

        // local = origin * motion   (both affine; origin row3 = [0,0,0,1])
        const float4* O4 = (const float4*)&sOrig[n * 16];
        float4 O0 = O4[0], O1 = O4[1], O2 = O4[2];   // 3x ds_load_b128
        float l00 = O0.x*r00 + O0.y*r10 + O0.z*r20;
        float l01 = O0.x*r01 + O0.y*r11 + O0.z*r21;
        float l02 = O0.x*r02 + O0.y*r12 + O0.z*r22;
        float l03 = O0.x*tx  + O0.y*ty  + O0.z*tz  + O0.w;
        float l10 = O1.x*r00 + O1.y*r10 + O1.z*r20;
        float l11 = O1.x*r01 + O1.y*r11 + O1.z*r21;
        float l12 = O1.x*r02 + O1.y*r12 + O1.z*r22;
        float l13 = O1.x*tx  + O1.y*ty  + O1.z*tz  + O1.w;
        float l20 = O2.x*r00 + O2.y*r10 + O2.z*r20;
        float l21 = O2.x*r01 + O2.y*r11 + O2.z*r21;
        float l22 = O2.x*r02 + O2.y*r12 + O2.z*r22;
        float l23 = O2.x*tx  + O2.y*ty  + O2.z*tz  + O2.w;

        // pose = pose * local
        float q00 = p00*l00 + p01*l10 + p02*l20;
        float q01 = p00*l01 + p01*l11 + p02*l21;
        float q02 = p00*l02 + p01*l12 + p02*l22;
        float q03 = p00*l03 + p01*l13 + p02*l23 + p03;
        float q10 = p10*l00 + p11*l10 + p12*l20;
        float q11 = p10*l01 + p11*l11 + p12*l21;
        float q12 = p10*l02 + p11*l12 + p12*l22;
        float q13 = p10*l03 + p11*l13 + p12*l23 + p13;
        float q20 = p20*l00 + p21*l10 + p22*l20;
        float q21 = p20*l01 + p21*l11 + p22*l21;
        float q22 = p20*l02 + p21*l12 + p22*l22;
        float q23 = p20*l03 + p21*l13 + p22*l23 + p23;
        p00=q00; p01=q01; p02=q02; p03=q03;
        p10=q10; p11=q11; p12=q12; p13=q13;
        p20=q20; p21=q21; p22=q22; p23=q23;

        // store 4x4 pose (row3 constant) with 128-bit stores
        float4* o4 = (float4*)(outB + (size_t)n * 16);
        o4[0] = make_float4(p00, p01, p02, p03);
        o4[1] = make_float4(p10, p11, p12, p13);
        o4[2] = make_float4(p20, p21, p22, p23);
        o4[3] = make_float4(0.f, 0.f, 0.f, 1.f);
    }
}

extern "C" void kernel_launch(void* const* d_in, const int* in_sizes, int n_in,
                              void* d_out, int out_size, void* d_ws, size_t ws_size,
                              hipStream_t stream) {
    (void)n_in; (void)d_ws; (void)ws_size; (void)out_size;
    const float* JA      = (const float*)d_in[0];
    const float* axes    = (const float*)d_in[1];
    const float* origins = (const float*)d_in[2];
    const float* mmul    = (const float*)d_in[3];
    const float* moff    = (const float*)d_in[4];
    const int*   ctrl    = (const int*)  d_in[5];
    const int*   mdst    = (const int*)  d_in[6];
    const int*   msrc    = (const int*)  d_in[7];
    const int*   jtype   = (const int*)  d_in[8];

    int nCtrl  = in_sizes[5];
    int nMimic = in_sizes[6];
    int nF     = in_sizes[8];
    int B      = in_sizes[0] / nCtrl;

    dim3 grid((B + BT - 1) / BT), block(BT);
    fk_chain_kernel<<<grid, block, 0, stream>>>(
        JA, axes, origins, mmul, moff, ctrl, mdst, msrc, jtype,
        (float*)d_out, B, nF, nCtrl, nMimic);
}